// BruteForceMoELinear_73693048865559
// MI455X (gfx1250) — compile-verified
//
#include <hip/hip_runtime.h>
#include <hip/hip_bf16.h>

// ---------------- constants ----------------
#define NUM_EXPERT 8
#define D_MODEL    512
#define D_FF       2048
#define N_TOKENS   2048
#define BATCH      4096          // N_TOKENS * TOP_K
#define FCHUNK     256           // F streamed in chunks of 256 columns
#define SAP        532           // padded LDS stride (dwords) for input tile (conflict-free, 16B aligned)
#define SHP        260           // padded LDS stride (dwords) for h4 chunk

typedef float v2f __attribute__((ext_vector_type(2)));
typedef float v8f __attribute__((ext_vector_type(8)));

static __device__ __forceinline__ v8f wmma_f32(v2f a, v2f b, v8f c) {
    // D = A(16x4 f32) * B(4x16 f32) + C(16x16 f32)
    return __builtin_amdgcn_wmma_f32_16x16x4_f32(
        /*neg_a=*/false, a, /*neg_b=*/false, b,
        /*c_mod=*/(short)0, c, /*reuse_a=*/false, /*reuse_b=*/false);
}

// ---------------- workspace layout (ints) ----------------
// wsI[0..8]    : offsets (row CSR, offsets[0]=0)
// wsI[16..23]  : counts -> then cursors
// wsI[32..40]  : tile offsets (prefix of ceil(cnt/16))
// wsI[64..64+BATCH) : row list grouped by expert

__global__ void moe_zero_counts(int* wsI) {
    if (threadIdx.x < 8) wsI[16 + threadIdx.x] = 0;
}

__global__ void moe_hist(const int* __restrict__ gidx, int* wsI) {
    int i = blockIdx.x * blockDim.x + threadIdx.x;
    if (i < BATCH) {
        int e = gidx[i] & 7;
        atomicAdd(&wsI[16 + e], 1);
    }
}

__global__ void moe_scan(int* wsI) {
    if (threadIdx.x == 0 && blockIdx.x == 0) {
        int off = 0, toff = 0;
        wsI[0] = 0; wsI[32] = 0;
        for (int e = 0; e < 8; ++e) {
            int c = wsI[16 + e];
            wsI[16 + e] = off;            // cursor for scatter
            off += c;
            wsI[1 + e] = off;             // offsets[e+1]
            toff += (c + 15) >> 4;
            wsI[33 + e] = toff;           // tileOff[e+1]
        }
    }
}

__global__ void moe_scatter(const int* __restrict__ gidx, int* wsI) {
    int i = blockIdx.x * blockDim.x + threadIdx.x;
    if (i < BATCH) {
        int e = gidx[i] & 7;
        int pos = atomicAdd(&wsI[16 + e], 1);
        wsI[64 + pos] = i;
    }
}

__global__ void moe_zero_out(float4* __restrict__ out4) {
    int i = blockIdx.x * blockDim.x + threadIdx.x;   // 1024*256 = 262144 = (N_TOKENS*D_MODEL)/4
    out4[i] = make_float4(0.f, 0.f, 0.f, 0.f);
}

// ---------------- main fused MoE kernel ----------------
// grid: 264 blocks (>= max total 16-row tiles), block: 256 threads (8 wave32)
__global__ __launch_bounds__(256) void moe_ffn_kernel(
    const float* __restrict__ inp,        // (BATCH, D_MODEL)
    const float* __restrict__ gate,       // (BATCH,) flat gate scores
    const float* __restrict__ w1,         // (E, D_FF, D_MODEL)
    const float* __restrict__ w2,         // (E, D_MODEL, D_FF)
    const int*   __restrict__ wsI,
    float*       __restrict__ out)        // (N_TOKENS, D_MODEL)
{
    __shared__ float sA[16 * SAP];        // gathered input tile 16 x 512 (padded)
    __shared__ float sH[16 * SHP];        // relu h4 chunk 16 x 256 (padded)
    __shared__ int   sRow[16];
    __shared__ float sGs[16];

    const int* offsets = wsI;
    const int* tileOff = wsI + 32;
    const int* rowlist = wsI + 64;

    int t = blockIdx.x;
    if (t >= tileOff[8]) return;
    int e = 0;
    while (e < 7 && t >= tileOff[e + 1]) ++e;
    int tloc     = t - tileOff[e];
    int rowStart = offsets[e] + tloc * 16;
    int nrows    = offsets[e + 1] - rowStart;
    if (nrows > 16) nrows = 16;

    const int tid = threadIdx.x;

    if (tid < 16) {
        int r = (tid < nrows) ? rowlist[rowStart + tid] : 0;
        sRow[tid] = r;
        sGs[tid]  = (tid < nrows) ? gate[r] : 0.0f;
    }

    // gather 16 rows x 512 f32 into LDS (float4 granularity: 2048 vec4, 8 per thread)
    #pragma unroll
    for (int i = 0; i < 8; ++i) {
        int idx = tid + i * 256;          // float4 index
        int m   = idx >> 7;               // 128 float4 per row
        int k4  = idx & 127;
        float4 v = make_float4(0.f, 0.f, 0.f, 0.f);
        if (m < nrows) {
            int r = rowlist[rowStart + m];
            v = *(const float4*)(inp + (size_t)r * D_MODEL + k4 * 4);
        }
        *(float4*)(&sA[m * SAP + k4 * 4]) = v;
    }
    __syncthreads();

    const int wave = tid >> 5;
    const int lane = tid & 31;
    const int mrow = lane & 15;
    const int koff = (lane >> 4) * 2;     // lanes 0-15 -> K0/K1, lanes 16-31 -> K2/K3
    const int rowB = (lane >> 4) * 8;     // C/D layout: M = g + 8*(lane>=16)

    const float* w1e = w1 + (size_t)e * ((size_t)D_FF * D_MODEL);
    const float* w2e = w2 + (size_t)e * ((size_t)D_MODEL * D_FF);

    v8f accY[4];
    #pragma unroll
    for (int i = 0; i < 4; ++i) accY[i] = (v8f){0,0,0,0,0,0,0,0};

    for (int c = 0; c < D_FF / FCHUNK; ++c) {
        const int fb = c * FCHUNK;

        // ---- phase A: h4 = relu(inp_tile @ W1^T) for this F chunk (2 N-tiles / wave)
        {
            int n0 = fb + wave * 32 + mrow;           // tile 0 column for this lane
            const float* pB0 = w1e + (size_t)n0 * D_MODEL + koff;
            const float* pB1 = pB0 + (size_t)16 * D_MODEL;
            const float* pA  = &sA[mrow * SAP + koff];
            v8f acc0 = (v8f){0,0,0,0,0,0,0,0};
            v8f acc1 = (v8f){0,0,0,0,0,0,0,0};
            #pragma unroll 4
            for (int k = 0; k < D_MODEL; k += 4) {
                v2f a  = *(const v2f*)(pA  + k);
                v2f b0 = *(const v2f*)(pB0 + k);
                v2f b1 = *(const v2f*)(pB1 + k);
                acc0 = wmma_f32(a, b0, acc0);
                acc1 = wmma_f32(a, b1, acc1);
            }
            int col0 = wave * 32 + (lane & 15);
            #pragma unroll
            for (int g = 0; g < 8; ++g) {
                float x0 = acc0[g]; x0 = x0 > 0.f ? x0 : 0.f;
                float x1 = acc1[g]; x1 = x1 > 0.f ? x1 : 0.f;
                sH[(rowB + g) * SHP + col0]      = x0;
                sH[(rowB + g) * SHP + col0 + 16] = x1;
            }
        }
        __syncthreads();

        // ---- phase B: y += h4_chunk @ W2^T  (4 D-tiles / wave, persistent accumulators)
        {
            const float* pA = &sH[mrow * SHP + koff];
            const float* pB[4];
            #pragma unroll
            for (int i = 0; i < 4; ++i) {
                int d = (wave + 8 * i) * 16 + (lane & 15);
                pB[i] = w2e + (size_t)d * D_FF + fb + koff;
            }
            #pragma unroll 4
            for (int k = 0; k < FCHUNK; k += 4) {
                v2f a = *(const v2f*)(pA + k);
                #pragma unroll
                for (int i = 0; i < 4; ++i) {
                    v2f b = *(const v2f*)(pB[i] + k);
                    accY[i] = wmma_f32(a, b, accY[i]);
                }
            }
        }
        __syncthreads();
    }

    // ---- gate-weighted combine into output (2 f32 addends per element -> deterministic)
    const int dl = lane & 15;
    #pragma unroll
    for (int i = 0; i < 4; ++i) {
        int d = (wave + 8 * i) * 16 + dl;
        #pragma unroll
        for (int g = 0; g < 8; ++g) {
            int m = rowB + g;
            if (m < nrows) {
                int   r   = sRow[m];
                float val = sGs[m] * accY[i][g];
                unsafeAtomicAdd(&out[(size_t)(r >> 1) * D_MODEL + d], val);
            }
        }
    }
}

// ---------------- launcher ----------------
extern "C" void kernel_launch(void* const* d_in, const int* in_sizes, int n_in,
                              void* d_out, int out_size, void* d_ws, size_t ws_size,
                              hipStream_t stream) {
    const float* inp   = (const float*)d_in[0];
    const int*   gidx  = (const int*)d_in[1];
    const float* gate  = (const float*)d_in[2];   // (N_TOKENS,1,TOP_K) flat == per-row score
    const float* w1    = (const float*)d_in[3];
    const float* w2    = (const float*)d_in[4];
    float*       out   = (float*)d_out;
    int*         wsI   = (int*)d_ws;

    moe_zero_counts<<<1, 32, 0, stream>>>(wsI);
    moe_hist<<<BATCH / 256, 256, 0, stream>>>(gidx, wsI);
    moe_scan<<<1, 32, 0, stream>>>(wsI);
    moe_scatter<<<BATCH / 256, 256, 0, stream>>>(gidx, wsI);
    moe_zero_out<<<1024, 256, 0, stream>>>((float4*)out);

    // max tiles = 4096/16 + (8-1) = 263
    moe_ffn_kernel<<<264, 256, 0, stream>>>(inp, gate, w1, w2, wsI, out);
}